// Measurement_35588099015224
// MI455X (gfx1250) — compile-verified
//
#include <hip/hip_runtime.h>

#define IN_DIM 9801
#define OUT_DIM 25
#define BS 16384

typedef __attribute__((ext_vector_type(2))) float v2f;
typedef __attribute__((ext_vector_type(8))) float v8f;

// ---------------------------------------------------------------------------
// Kernel 1: zero the 50-float (idx|val) table in workspace (ws is poisoned).
// ---------------------------------------------------------------------------
__global__ __launch_bounds__(64) void zero_tab(float* __restrict__ tab) {
    if (threadIdx.x < 2 * OUT_DIM) tab[threadIdx.x] = 0.0f;
}

// ---------------------------------------------------------------------------
// Kernel 2: recover (index, value) per measure row as a real f32 WMMA GEMM:
//   [idx_f | val] = measure(25x9801) @ [iota | ones](9801x2)
// K-dimension split across blocks; partials combined with f32 atomics
// (exact: every partial but one is 0.0).
//
// V_WMMA_F32_16X16X4_F32 layouts (wave32):
//   A 16x4 : M = lane&15 ; VGPR0 -> K = 2h, VGPR1 -> K = 2h+1  (h = lane>>4)
//   B 4x16 : N = lane&15 ; VGPR v, half h -> K = v + 2h
//   C 16x16: N = lane&15 ; VGPR r -> M = r + 8h
// ---------------------------------------------------------------------------
__global__ __launch_bounds__(64) void extract_idx_wmma(const float* __restrict__ measure,
                                                       float* __restrict__ tab,
                                                       int k_chunk) {
    const int lane  = threadIdx.x & 31;
    const int wave  = threadIdx.x >> 5;      // wave 0 -> rows 0..15, wave 1 -> rows 16..31
    const int mbase = wave * 16;
    const int m     = mbase + (lane & 15);
    const int h     = lane >> 4;             // lane-half -> K pair selector
    const int n     = lane & 15;             // B/C column

    const float rowmask = (m < OUT_DIM) ? 1.0f : 0.0f;
    const int   rowc    = (m < OUT_DIM) ? m : 0;          // clamped row (safe loads)
    const int   rowoff  = rowc * IN_DIM;

    const int k_begin = blockIdx.x * k_chunk;
    int k_end = k_begin + k_chunk;
    if (k_end > IN_DIM) k_end = IN_DIM;

    v8f c = {};
    for (int k0 = k_begin; k0 < k_end; k0 += 4) {
        const int ka = k0 + 2 * h;           // K for A.x / B.x
        const int kb = ka + 1;               // K for A.y / B.y
        const float pa = (ka < k_end) ? 1.0f : 0.0f;   // in-chunk & in-tensor guard
        const float pb = (kb < k_end) ? 1.0f : 0.0f;
        const int kca = (ka < IN_DIM) ? ka : (IN_DIM - 1);
        const int kcb = (kb < IN_DIM) ? kb : (IN_DIM - 1);

        v2f a;
        a.x = measure[rowoff + kca] * (pa * rowmask);
        a.y = measure[rowoff + kcb] * (pb * rowmask);

        v2f b;
        b.x = pa * ((n == 0) ? (float)ka : ((n == 1) ? 1.0f : 0.0f));
        b.y = pb * ((n == 0) ? (float)kb : ((n == 1) ? 1.0f : 0.0f));

        // (neg_a, A, neg_b, B, c_mod, C, reuse_a, reuse_b)
        c = __builtin_amdgcn_wmma_f32_16x16x4_f32(false, a, false, b,
                                                  (short)0, c, false, false);
    }

    // Columns 0 (idx) and 1 (val) live in lanes 0,1 (rows mbase+r) and
    // lanes 16,17 (rows mbase+8+r).
    if (n < 2) {
        #pragma unroll
        for (int r = 0; r < 8; ++r) {
            const int row = mbase + r + 8 * h;
            if (row < OUT_DIM) atomicAdd(&tab[n * OUT_DIM + row], c[r]);
        }
    }
}

// ---------------------------------------------------------------------------
// Kernel 3: out[b,o] = input[b, idx[o]] * val[o].
// Coalesced NT stores; scattered read-once NT loads (don't pollute 192MB L2).
// ---------------------------------------------------------------------------
__global__ __launch_bounds__(256) void gather_out(const float* __restrict__ in,
                                                  const float* __restrict__ tab,
                                                  float* __restrict__ out) {
    __shared__ int   s_idx[OUT_DIM];
    __shared__ float s_val[OUT_DIM];
    if (threadIdx.x < OUT_DIM) {
        const float fi = tab[threadIdx.x];           // exact integer in f32
        s_idx[threadIdx.x] = (int)(fi + 0.5f);
        s_val[threadIdx.x] = tab[OUT_DIM + threadIdx.x];
    }
    __syncthreads();

    const int t = blockIdx.x * 256 + threadIdx.x;    // [0, BS*OUT_DIM)
    if (t < BS * OUT_DIM) {
        const int b = t / OUT_DIM;
        const int o = t - b * OUT_DIM;
        const float v = __builtin_nontemporal_load(&in[b * IN_DIM + s_idx[o]]) * s_val[o];
        __builtin_nontemporal_store(v, &out[t]);
    }
}

extern "C" void kernel_launch(void* const* d_in, const int* in_sizes, int n_in,
                              void* d_out, int out_size, void* d_ws, size_t ws_size,
                              hipStream_t stream) {
    const float* x       = (const float*)d_in[0];   // [BS, IN_DIM]
    const float* measure = (const float*)d_in[1];   // [OUT_DIM, IN_DIM]
    float*       out     = (float*)d_out;           // [BS, OUT_DIM]
    float*       tab     = (float*)d_ws;            // 50 floats: idx[25] | val[25]

    zero_tab<<<1, 64, 0, stream>>>(tab);

    const int k_chunk = 100;                         // multiple of 4
    const int nblk    = (IN_DIM + k_chunk - 1) / k_chunk;   // 99 blocks
    extract_idx_wmma<<<nblk, 64, 0, stream>>>(measure, tab, k_chunk);

    const int total = BS * OUT_DIM;                  // 409,600
    gather_out<<<(total + 255) / 256, 256, 0, stream>>>(x, tab, out);
}